// PointToSurfaceLoss_64338610094477
// MI455X (gfx1250) — compile-verified
//
#include <hip/hip_runtime.h>
#include <math.h>

// CDNA5 wave32 WMMA vector types
typedef __attribute__((ext_vector_type(2))) float v2f;   // A/B of 16x16x4 f32: 64 f32 / 32 lanes
typedef __attribute__((ext_vector_type(8))) float v8f;   // C/D 16x16 f32

#define EPS_DIV  1e-12f
#define EPS_BARY 1e-16f
#define EPS_GMO  1e-16f

// ---------------------------------------------------------------------------
// Kernel A: gather triangles + precompute per-face constants
//   K0 = (ab.a, ac.a, ab.b, ac.b)   K1 = (ab.c, ac.c, |a|^2, |b|^2)
//   K2 = (|c|^2, 0, 0, 0)
// ---------------------------------------------------------------------------
__global__ void precompute_faces(const float* __restrict__ verts, // [B,V,3]
                                 const int* __restrict__ faces,   // [F,3]
                                 float4* __restrict__ triA,
                                 float4* __restrict__ triB,
                                 float4* __restrict__ triC,
                                 float4* __restrict__ K,          // [B*F*3]
                                 int Bn, int V, int F)
{
    int idx = blockIdx.x * blockDim.x + threadIdx.x;
    if (idx >= Bn * F) return;
    int b = idx / F, f = idx % F;
    int i0 = faces[3*f+0], i1 = faces[3*f+1], i2 = faces[3*f+2];
    const float* vb = verts + (size_t)b * V * 3;
    float ax = vb[3*i0+0], ay = vb[3*i0+1], az = vb[3*i0+2];
    float bx = vb[3*i1+0], by = vb[3*i1+1], bz = vb[3*i1+2];
    float cx = vb[3*i2+0], cy = vb[3*i2+1], cz = vb[3*i2+2];
    triA[idx] = make_float4(ax, ay, az, 0.f);
    triB[idx] = make_float4(bx, by, bz, 0.f);
    triC[idx] = make_float4(cx, cy, cz, 0.f);
    float abx = bx-ax, aby = by-ay, abz = bz-az;
    float acx = cx-ax, acy = cy-ay, acz = cz-az;
    float aba = abx*ax + aby*ay + abz*az;
    float aca = acx*ax + acy*ay + acz*az;
    float abb = abx*bx + aby*by + abz*bz;
    float acb = acx*bx + acy*by + acz*bz;
    float abc = abx*cx + aby*cy + abz*cz;
    float acc = acx*cx + acy*cy + acz*cz;
    float na = ax*ax + ay*ay + az*az;
    float nb = bx*bx + by*by + bz*bz;
    float nc = cx*cx + cy*cy + cz*cz;
    K[(size_t)idx*3+0] = make_float4(aba, aca, abb, acb);
    K[(size_t)idx*3+1] = make_float4(abc, acc, na, nb);
    K[(size_t)idx*3+2] = make_float4(nc, 0.f, 0.f, 0.f);
}

// ---------------------------------------------------------------------------
// Kernel B: one wave32 per 16 points; WMMA f32 16x16x4 computes the Gram dots
//   D_a[r][lane] = p_M . a_face   (M = r + 8*(lane/16), face = tile*16 + lane%16)
// then a fully BRANCHLESS per-lane region classification + running argmin.
// ---------------------------------------------------------------------------
__global__ void __launch_bounds__(128)
nearest_face(const float* __restrict__ points, // [B,P,3]
             const float4* __restrict__ triA,
             const float4* __restrict__ triB,
             const float4* __restrict__ triC,
             const float4* __restrict__ K,
             int* __restrict__ faceIdx,        // [B*P]
             int Bn, int P, int F)
{
    const int lane = threadIdx.x & 31;
    const int wave = (blockIdx.x * blockDim.x + threadIdx.x) >> 5;
    const int wavesPerBatch = P >> 4;
    const int b   = wave / wavesPerBatch;
    const int pt0 = (wave % wavesPerBatch) << 4;
    const int half = lane >> 4;        // K half: lanes 0-15 -> K0,1 ; 16-31 -> K2,3
    const int col  = lane & 15;

    const float* ppt = points + ((size_t)b * P + pt0) * 3;

    // A matrix: lane holds point (pt0+col); (x,y) in half 0, (z,0) in half 1.
    v2f Amat;
    {
        const float* pp = ppt + col * 3;
        Amat[0] = half ? pp[2] : pp[0];
        Amat[1] = half ? 0.f   : pp[1];
    }
    // |p|^2 for the 8 D-rows this lane sees (M = r + 8*half)
    float pn[8];
    #pragma unroll
    for (int r = 0; r < 8; ++r) {
        const float* pp = ppt + (r + 8*half) * 3;
        pn[r] = pp[0]*pp[0] + pp[1]*pp[1] + pp[2]*pp[2];
    }

    float best[8]; int bidx[8];
    #pragma unroll
    for (int r = 0; r < 8; ++r) { best[r] = 3.4e38f; bidx[r] = 0; }

    const int nTiles = F >> 4;
    const size_t fbase0 = (size_t)b * F;
    for (int t = 0; t < nTiles; ++t) {
        const int fl = (t << 4) + col;
        const size_t fb = fbase0 + fl;
        float4 a4 = triA[fb], b4 = triB[fb], c4 = triC[fb];
        v2f Ba, Bb, Bc;
        Ba[0] = half ? a4.z : a4.x;  Ba[1] = half ? 0.f : a4.y;
        Bb[0] = half ? b4.z : b4.x;  Bb[1] = half ? 0.f : b4.y;
        Bc[0] = half ? c4.z : c4.x;  Bc[1] = half ? 0.f : c4.y;
        v8f zero = {0.f,0.f,0.f,0.f,0.f,0.f,0.f,0.f};
        v8f Da = __builtin_amdgcn_wmma_f32_16x16x4_f32(false, Amat, false, Ba,
                                                       (short)0, zero, false, false);
        v8f Db = __builtin_amdgcn_wmma_f32_16x16x4_f32(false, Amat, false, Bb,
                                                       (short)0, zero, false, false);
        v8f Dc = __builtin_amdgcn_wmma_f32_16x16x4_f32(false, Amat, false, Bc,
                                                       (short)0, zero, false, false);
        float4 k0 = K[fb*3+0], k1 = K[fb*3+1], k2 = K[fb*3+2];
        const float abac = k0.w - k0.y;            // ab . ac

        #pragma unroll
        for (int r = 0; r < 8; ++r) {
            float pa = Da[r], pb = Db[r], pc = Dc[r];
            float u  = pb - pa;          // ab . p
            float tt = pc - pa;          // ac . p
            float d1 = u  - k0.x, d2 = tt - k0.y;
            float d3 = u  - k0.z, d4 = tt - k0.w;
            float d5 = u  - k1.x, d6 = tt - k1.y;
            float apn = fmaf(-2.f, pa, pn[r]) + k1.z;   // |p-a|^2
            float bpn = fmaf(-2.f, pb, pn[r]) + k1.w;   // |p-b|^2
            float cpn = fmaf(-2.f, pc, pn[r]) + k2.x;   // |p-c|^2
            float va  = d3*d6 - d5*d4;
            float vbv = d5*d2 - d1*d6;
            float vcv = d1*d4 - d3*d2;
            float ab2  = d1 - d3;   // |ab|^2
            float ac2  = d2 - d6;   // |ac|^2
            float bcbp = d4 - d3;   // bc . (p-b)
            float bccp = d5 - d6;   // bc . (c-p)

            // Branchless select chain, same precedence as the reference's
            // jnp.where stack: interior < bc < ac < ab < c < b < a.

            // interior (lowest priority)
            float rden = 1.f / fmaxf(va + vbv + vcv, EPS_DIV);
            float vv = vbv * rden, ww = vcv * rden;
            float dist = apn - 2.f*(vv*d1 + ww*d2)
                       + vv*vv*ab2 + 2.f*vv*ww*abac + ww*ww*ac2;

            // edge BC
            float bc2 = bcbp + bccp;
            float tbc = bcbp / fmaxf(bc2, EPS_DIV);
            float dBC = bpn - tbc*(2.f*bcbp - tbc*bc2);
            bool mBC = (va <= 0.f) & (bcbp >= 0.f) & (bccp >= 0.f);
            dist = mBC ? dBC : dist;

            // edge AC
            float tac = d2 / fmaxf(ac2, EPS_DIV);
            float dAC = apn - tac*(2.f*d2 - tac*ac2);
            bool mAC = (vbv <= 0.f) & (d2 >= 0.f) & (d6 <= 0.f);
            dist = mAC ? dAC : dist;

            // edge AB
            float tab = d1 / fmaxf(ab2, EPS_DIV);
            float dAB = apn - tab*(2.f*d1 - tab*ab2);
            bool mAB = (vcv <= 0.f) & (d1 >= 0.f) & (d3 <= 0.f);
            dist = mAB ? dAB : dist;

            // vertex regions (highest priority, a last)
            bool mC = (d6 >= 0.f) & (d5 <= d6);
            dist = mC ? cpn : dist;
            bool mB = (d3 >= 0.f) & (d4 <= d3);
            dist = mB ? bpn : dist;
            bool mA = (d1 <= 0.f) & (d2 <= 0.f);
            dist = mA ? apn : dist;

            bool upd = dist < best[r];           // strict: keeps first (lowest
            best[r] = upd ? dist : best[r];      // face index) within a lane
            bidx[r] = upd ? fl   : bidx[r];
        }
    }

    // cross-lane argmin over the 16 faces/lanes of each half,
    // first-occurrence tie-break (smallest face index on equal distance)
    #pragma unroll
    for (int r = 0; r < 8; ++r) {
        float d = best[r]; int i = bidx[r];
        #pragma unroll
        for (int m = 1; m < 16; m <<= 1) {
            float od = __shfl_xor(d, m, 32);
            int   oi = __shfl_xor(i, m, 32);
            bool take = (od < d) | ((od == d) & (oi < i));
            d = take ? od : d;
            i = take ? oi : i;
        }
        if (col == 0) faceIdx[(size_t)b * P + pt0 + r + 8*half] = i;
    }
}

// ---------------------------------------------------------------------------
// Kernel C: exact barycentric projection + GMO + mean (reference formulas)
// ---------------------------------------------------------------------------
__global__ void finalize(const float* __restrict__ points,
                         const float4* __restrict__ triA,
                         const float4* __restrict__ triB,
                         const float4* __restrict__ triC,
                         const int* __restrict__ faceIdx,
                         float* __restrict__ out,
                         int Bn, int P, int F)
{
    __shared__ float sdata[256];
    float acc = 0.f;
    const int total = Bn * P;
    for (int i = threadIdx.x; i < total; i += 256) {
        int b = i / P;
        const float* q = points + (size_t)i * 3;
        int fi = faceIdx[i];
        size_t fb = (size_t)b * F + fi;
        float4 a4 = triA[fb], b4 = triB[fb], c4 = triC[fb];
        float v1x = b4.x-a4.x, v1y = b4.y-a4.y, v1z = b4.z-a4.z;
        float v2x = c4.x-a4.x, v2y = c4.y-a4.y, v2z = c4.z-a4.z;
        float nx = v1y*v2z - v1z*v2y;
        float ny = v1z*v2x - v1x*v2z;
        float nz = v1x*v2y - v1y*v2x;
        float ndn = nx*nx + ny*ny + nz*nz;
        ndn = (ndn < EPS_BARY) ? 1.f : ndn;
        float wx = q[0]-a4.x, wy = q[1]-a4.y, wz = q[2]-a4.z;
        float c1x = v1y*wz - v1z*wy, c1y = v1z*wx - v1x*wz, c1z = v1x*wy - v1y*wx;
        float gamma = (c1x*nx + c1y*ny + c1z*nz) / ndn;
        float c2x = wy*v2z - wz*v2y, c2y = wz*v2x - wx*v2z, c2z = wx*v2y - wy*v2x;
        float beta  = (c2x*nx + c2y*ny + c2z*nz) / ndn;
        float alpha = 1.f - beta - gamma;
        float cpx = a4.x*alpha + b4.x*beta + c4.x*gamma;
        float cpy = a4.y*alpha + b4.y*beta + c4.y*gamma;
        float cpz = a4.z*alpha + b4.z*beta + c4.z*gamma;
        float dx = cpx - q[0], dy = cpy - q[1], dz = cpz - q[2];
        float sq = dx*dx + dy*dy + dz*dz;
        acc += sqrtf(sq / (1.f + sq) + EPS_GMO);   // sigma = 1
    }
    sdata[threadIdx.x] = acc;
    __syncthreads();
    for (int s = 128; s > 0; s >>= 1) {
        if (threadIdx.x < s) sdata[threadIdx.x] += sdata[threadIdx.x + s];
        __syncthreads();
    }
    if (threadIdx.x == 0) out[0] = sdata[0] / (float)total;
}

// ---------------------------------------------------------------------------
extern "C" void kernel_launch(void* const* d_in, const int* in_sizes, int n_in,
                              void* d_out, int out_size, void* d_ws, size_t ws_size,
                              hipStream_t stream)
{
    const float* points = (const float*)d_in[0];   // [B,P,3] f32
    const float* verts  = (const float*)d_in[1];   // [B,V,3] f32
    const int*   faces  = (const int*)d_in[2];     // [F,3] i32

    const int Bn = 2;
    const int P  = in_sizes[0] / (Bn * 3);         // 2048
    const int V  = in_sizes[1] / (Bn * 3);         // 2562
    const int F  = in_sizes[2] / 3;                // 5120

    char* ws = (char*)d_ws;
    float4* triA = (float4*)ws;  ws += (size_t)Bn * F * sizeof(float4);
    float4* triB = (float4*)ws;  ws += (size_t)Bn * F * sizeof(float4);
    float4* triC = (float4*)ws;  ws += (size_t)Bn * F * sizeof(float4);
    float4* Kc   = (float4*)ws;  ws += (size_t)Bn * F * 3 * sizeof(float4);
    int* faceIdx = (int*)ws;

    const int nBF = Bn * F;
    precompute_faces<<<(nBF + 255) / 256, 256, 0, stream>>>(
        verts, faces, triA, triB, triC, Kc, Bn, V, F);

    const int nWaves = Bn * (P / 16);              // 256 waves, exact grid
    nearest_face<<<(nWaves * 32) / 128, 128, 0, stream>>>(
        points, triA, triB, triC, Kc, faceIdx, Bn, P, F);

    finalize<<<1, 256, 0, stream>>>(
        points, triA, triB, triC, faceIdx, (float*)d_out, Bn, P, F);
}